// HDSR_55224689492112
// MI455X (gfx1250) — compile-verified
//
#include <hip/hip_runtime.h>
#include <hip/hip_bf16.h>
#include <stdint.h>

// ---------------------------------------------------------------------------
// HDSR soc/perturbed propagation: 3 x (edge-dropout -> scatter-softmax ->
// LGConv) -> layer mean.  COO scatter formulation.
//   - native global_atomic_add_f32 scatters (L2-resident working set)
//   - NT temporal hints on the read-once edge stream to protect L2 residency
//   - wave32 half-wave leader computes per-edge coef, broadcast via __shfl
// Inputs: emb[f32 N*64], edge_attr[f32 E], edge_index[i32 2*E],
//         edge_mask[u8 L*E].  Output: f32 N*64.
// ---------------------------------------------------------------------------

__global__ void hdsr_init(const float* __restrict__ emb,
                          float* __restrict__ x_cur,
                          float* __restrict__ out,
                          int n_feat) {
    int i = blockIdx.x * blockDim.x + threadIdx.x;
    if (i < n_feat) {
        float v = __builtin_nontemporal_load(emb + i);
        x_cur[i] = v;                              // x^(0): keep resident (gathered)
        __builtin_nontemporal_store(v, out + i);   // acc = emb: streaming
    }
}

__global__ void hdsr_zero_layer(float* __restrict__ x_nxt,
                                unsigned* __restrict__ mB,
                                float* __restrict__ sB,
                                int n_feat, int n_nodes) {
    int i = blockIdx.x * blockDim.x + threadIdx.x;
    if (i < n_feat) x_nxt[i] = 0.0f;
    if (i < n_nodes) { mB[i] = 0u; sB[i] = 0.0f; }
}

// segment max of logits over dst.  edge_attr >= 0, so the IEEE bit pattern is
// monotone under unsigned compare; init 0 reproduces m_safe=0 for empty groups.
__global__ void hdsr_edge_max(const unsigned char* __restrict__ mask,
                              const int* __restrict__ dsti,
                              const float* __restrict__ w,
                              unsigned* __restrict__ mB,
                              int n_edges) {
    int e = blockIdx.x * blockDim.x + threadIdx.x;
    if (e >= n_edges) return;
    if (!__builtin_nontemporal_load(mask + e)) return;
    int dv = __builtin_nontemporal_load(dsti + e);
    atomicMax(&mB[dv], __float_as_uint(__builtin_nontemporal_load(w + e)));
}

// segment sum of exp(w - m[dst]) over dst (softmax denominator).
__global__ void hdsr_edge_sum(const unsigned char* __restrict__ mask,
                              const int* __restrict__ dsti,
                              const float* __restrict__ w,
                              const unsigned* __restrict__ mB,
                              float* __restrict__ sB,
                              int n_edges) {
    int e = blockIdx.x * blockDim.x + threadIdx.x;
    if (e >= n_edges) return;
    if (!__builtin_nontemporal_load(mask + e)) return;
    int dv = __builtin_nontemporal_load(dsti + e);
    float we = __builtin_nontemporal_load(w + e);
    unsafeAtomicAdd(&sB[dv], expf(we - __uint_as_float(mB[dv])));
}

// Weighted propagation: x_nxt[dst] += dis[src] * (e/s[dst]) * x_cur[src].
// 16 lanes per edge (2 edges per wave32), float4 per lane -> one
// global_load_b128 gather and four global_atomic_add_f32 per lane.
// Half-wave leaders (lanes 0,16) compute coef once; __shfl broadcast.
// dis[dst]==1 for any retained edge (its own exp term makes s[dst]>0);
// dis[src] = (s[src]>0) ? 1 : 0.
__global__ void hdsr_edge_prop(const unsigned char* __restrict__ mask,
                               const int* __restrict__ srci,
                               const int* __restrict__ dsti,
                               const float* __restrict__ w,
                               const unsigned* __restrict__ mB,
                               const float* __restrict__ sB,
                               const float* __restrict__ x_cur,
                               float* __restrict__ x_nxt,
                               int n_edges) {
    long long t = (long long)blockIdx.x * blockDim.x + threadIdx.x;
    int e    = (int)(t >> 4);
    int lane = threadIdx.x & 31;
    int lead = lane & 16;          // leader lane of this half-wave
    int q    = lane & 15;          // float4 slot within the 64-dim row

    float coef = 0.0f;
    int sv = 0, dv = 0;
    if ((lane & 15) == 0 && e < n_edges &&
        __builtin_nontemporal_load(mask + e)) {
        sv = __builtin_nontemporal_load(srci + e);
        dv = __builtin_nontemporal_load(dsti + e);
        float we   = __builtin_nontemporal_load(w + e);
        float sd   = sB[dv];
        float diss = (sB[sv] > 0.0f) ? 1.0f : 0.0f;
        coef = diss * expf(we - __uint_as_float(mB[dv])) / fmaxf(sd, 1e-16f);
    }
    coef = __shfl(coef, lead);
    sv   = __shfl(sv,   lead);
    dv   = __shfl(dv,   lead);
    if (coef == 0.0f) return;      // dropped edge / OOB / dis_src==0: contributes 0

    const float4 xv = *(const float4*)(x_cur + (size_t)sv * 64 + (size_t)q * 4);
    float* op = x_nxt + (size_t)dv * 64 + (size_t)q * 4;
    unsafeAtomicAdd(op + 0, coef * xv.x);
    unsafeAtomicAdd(op + 1, coef * xv.y);
    unsafeAtomicAdd(op + 2, coef * xv.z);
    unsafeAtomicAdd(op + 3, coef * xv.w);
}

// acc += x ; on the last layer also apply the 1/(L+1) mean.
// acc stream is read/written once per layer -> NT; x stays RT (gathered next).
__global__ void hdsr_accum(float* __restrict__ out,
                           const float* __restrict__ x,
                           float scale, int n_feat) {
    int i = blockIdx.x * blockDim.x + threadIdx.x;
    if (i < n_feat) {
        float v = (__builtin_nontemporal_load(out + i) + x[i]) * scale;
        __builtin_nontemporal_store(v, out + i);
    }
}

extern "C" void kernel_launch(void* const* d_in, const int* in_sizes, int n_in,
                              void* d_out, int out_size, void* d_ws, size_t ws_size,
                              hipStream_t stream) {
    const float*         emb     = (const float*)d_in[0];
    const float*         w       = (const float*)d_in[1];
    const int*           ei      = (const int*)d_in[2];
    const unsigned char* maskall = (const unsigned char*)d_in[3];
    float*               out     = (float*)d_out;

    const int n_feat   = in_sizes[0];          // n_nodes * 64
    const int n_nodes  = n_feat / 64;
    const int n_edges  = in_sizes[1];
    const int n_layers = in_sizes[3] / n_edges;

    const int* srci = ei;
    const int* dsti = ei + n_edges;

    // workspace layout: x_cur | x_nxt | m | s   (~78 MB)
    float*    x_cur = (float*)d_ws;
    float*    x_nxt = x_cur + (size_t)n_feat;
    unsigned* mB    = (unsigned*)(x_nxt + (size_t)n_feat);
    float*    sB    = (float*)(mB + n_nodes);

    const int B  = 256;
    const int gF = (n_feat + B - 1) / B;
    const int gE = (n_edges + B - 1) / B;
    const int gP = (int)(((long long)n_edges * 16 + B - 1) / B);

    hdsr_init<<<gF, B, 0, stream>>>(emb, x_cur, out, n_feat);

    for (int l = 0; l < n_layers; ++l) {
        const unsigned char* mask = maskall + (size_t)l * n_edges;
        hdsr_zero_layer<<<gF, B, 0, stream>>>(x_nxt, mB, sB, n_feat, n_nodes);
        hdsr_edge_max <<<gE, B, 0, stream>>>(mask, dsti, w, mB, n_edges);
        hdsr_edge_sum <<<gE, B, 0, stream>>>(mask, dsti, w, mB, sB, n_edges);
        hdsr_edge_prop<<<gP, B, 0, stream>>>(mask, srci, dsti, w, mB, sB,
                                             x_cur, x_nxt, n_edges);
        const float sc = (l == n_layers - 1) ? 1.0f / (float)(n_layers + 1) : 1.0f;
        hdsr_accum    <<<gF, B, 0, stream>>>(out, x_nxt, sc, n_feat);
        float* tmp = x_cur; x_cur = x_nxt; x_nxt = tmp;
    }
}